// ChannelIndependentConv_51118700757133
// MI455X (gfx1250) — compile-verified
//
#include <hip/hip_runtime.h>

typedef float v2f __attribute__((ext_vector_type(2)));
typedef float v8f __attribute__((ext_vector_type(8)));

#define BB 8
#define NN 256
#define DD 64
#define LSTRIDE 68   // padded LDS row stride (floats): keeps float4 alignment, kills bank conflicts

__device__ __forceinline__ v8f wmma_f32_4(v2f a, v2f b, v8f c) {
  // D = A(16x4,f32) x B(4x16,f32) + C(16x16,f32)
  return __builtin_amdgcn_wmma_f32_16x16x4_f32(false, a, false, b, (short)0, c, false, false);
}

// ---------------------------------------------------------------------------
// Kernel 1: node_x = emb_node @ Wn + bn        (raw, stored to ws)
//           relu_sx = relu(emb_node @ Wsn+bsn) (stored to ws)
// Block: 256 threads (8 waves) handles 64 rows. Grid: 2048/64 = 32 blocks.
// ---------------------------------------------------------------------------
__global__ __launch_bounds__(256) void node_gemm_kernel(
    const float* __restrict__ emb_node, const float* __restrict__ Wn,
    const float* __restrict__ bn, const float* __restrict__ Wsn,
    const float* __restrict__ bsn, float* __restrict__ node_x,
    float* __restrict__ relu_sx) {
  __shared__ __align__(16) float sE[64 * LSTRIDE];
  __shared__ __align__(16) float sWnT[64 * LSTRIDE];
  __shared__ __align__(16) float sWsT[64 * LSTRIDE];
  __shared__ float sb[2][64];

  const int tid = threadIdx.x;
  const int r0 = blockIdx.x * 64;

  // Stage 64 emb_node rows, float4 coalesced, into padded LDS [r][k]
#pragma unroll
  for (int e = 0; e < 4; ++e) {
    int idx = (tid + e * 256) * 4;              // element index 0..4095
    int r = idx >> 6, k = idx & 63;
    *(float4*)&sE[r * LSTRIDE + k] = *(const float4*)(emb_node + (size_t)r0 * DD + idx);
  }
  // Stage both weights transposed: sW?T[c][k] = W[k][c]
#pragma unroll
  for (int e = 0; e < 16; ++e) {
    int idx = tid + e * 256;                    // 0..4095
    int k = idx >> 6, c = idx & 63;
    sWnT[c * LSTRIDE + k] = Wn[idx];
    sWsT[c * LSTRIDE + k] = Wsn[idx];
  }
  if (tid < 64) { sb[0][tid] = bn[tid]; sb[1][tid] = bsn[tid]; }
  __syncthreads();

  const int wave = tid >> 5, lane = tid & 31;
  const int half = lane >> 4, lidx = lane & 15;

  // 2 GEMMs x 16 tiles = 32 tiles; 8 waves x 4 tiles each
#pragma unroll
  for (int t = 0; t < 4; ++t) {
    int tileid = wave * 4 + t;                  // 0..31
    int g = tileid >> 4;                        // 0: Wn, 1: Wsn
    int mt = (tileid >> 2) & 3, nt = tileid & 3;
    const float* sW = g ? sWsT : sWnT;
    int c = nt * 16 + lidx;
    float bias = sb[g][c];
    v8f acc;
#pragma unroll
    for (int v = 0; v < 8; ++v) acc[v] = bias;
#pragma unroll
    for (int kk = 0; kk < 16; ++kk) {
      int k0 = kk * 4 + 2 * half;
      v2f a  = *(const v2f*)&sE[(mt * 16 + lidx) * LSTRIDE + k0];
      v2f bf = *(const v2f*)&sW[c * LSTRIDE + k0];
      acc = wmma_f32_4(a, bf, acc);
    }
    float* dst = g ? relu_sx : node_x;
#pragma unroll
    for (int v = 0; v < 8; ++v) {
      int r = r0 + mt * 16 + v + 8 * half;      // D layout: VGPR v -> M = v (+8 for hi half)
      float val = acc[v];
      if (g) val = fmaxf(val, 0.f);
      dst[(size_t)r * DD + c] = val;
    }
  }
}

// ---------------------------------------------------------------------------
// Kernel 2: per (b,i) row block:
//   edge_x = emb_edge[b,i] @ We + be        (WMMA f32, 4 chunks of 64 j-rows)
//   edge_out = relu(edge_x)                 (LDS write-back, coalesced B128 store)
//   agg[c] = sum_j A[b,i,j]*edge_x[j,c]*node_x[b,j,c]  (ds_add_f32 atomics)
//   node_out = relu(agg) + relu_sx
// Grid: 2048 blocks x 256 threads.
// ---------------------------------------------------------------------------
__global__ __launch_bounds__(256) void edge_kernel(
    const float* __restrict__ Amat, const float* __restrict__ emb_edge,
    const float* __restrict__ We, const float* __restrict__ be,
    const float* __restrict__ node_x, const float* __restrict__ relu_sx,
    float* __restrict__ node_out, float* __restrict__ edge_out) {
  __shared__ __align__(16) float sWeT[64 * LSTRIDE];  // We transposed: [c][k]
  __shared__ __align__(16) float sE[64 * LSTRIDE];    // edge chunk [j][k], reused as relu [j][c]
  __shared__ __align__(16) float sN[64 * LSTRIDE];    // node_x chunk [j][c]
  __shared__ float sMask[64];
  __shared__ float sAgg[64];
  __shared__ float sbe[64];

  const int tid = threadIdx.x;
  const int bi = blockIdx.x;                  // = b*256 + i
  const int b = bi >> 8;
  const size_t rowbase = (size_t)bi * (NN * DD);

#pragma unroll
  for (int e = 0; e < 16; ++e) {
    int idx = tid + e * 256;
    sWeT[(idx & 63) * LSTRIDE + (idx >> 6)] = We[idx];
  }
  if (tid < 64) { sbe[tid] = be[tid]; sAgg[tid] = 0.f; }
  __syncthreads();

  const int wave = tid >> 5, lane = tid & 31;
  const int half = lane >> 4, lidx = lane & 15;
  float rstash[2][8];

  for (int chunk = 0; chunk < 4; ++chunk) {
    const int jbase = chunk * 64;
    // Stage 64 edge rows + 64 node_x rows (float4, coalesced)
#pragma unroll
    for (int e = 0; e < 4; ++e) {
      int idx = (tid + e * 256) * 4;
      int j = idx >> 6, k = idx & 63;
      *(float4*)&sE[j * LSTRIDE + k] =
          *(const float4*)(emb_edge + rowbase + (size_t)jbase * DD + idx);
      *(float4*)&sN[j * LSTRIDE + k] =
          *(const float4*)(node_x + ((size_t)b * NN + jbase) * DD + idx);
    }
    if (tid < 64) sMask[tid] = Amat[(size_t)bi * NN + jbase + tid];
    // Warm L2/L0 with next chunk (global_prefetch_b8), one cacheline per lane
    if (chunk < 3)
      __builtin_prefetch(emb_edge + rowbase + (size_t)(jbase + 64) * DD + tid * 16, 0, 1);
    __syncthreads();

    // 16 output tiles (4 mt x 4 nt); 2 per wave
#pragma unroll
    for (int t = 0; t < 2; ++t) {
      int tile = wave * 2 + t;
      int mt = tile >> 2, nt = tile & 3;
      int c = nt * 16 + lidx;
      float bias = sbe[c];
      v8f acc;
#pragma unroll
      for (int v = 0; v < 8; ++v) acc[v] = bias;
#pragma unroll
      for (int kk = 0; kk < 16; ++kk) {
        int k0 = kk * 4 + 2 * half;
        v2f a  = *(const v2f*)&sE[(mt * 16 + lidx) * LSTRIDE + k0];
        v2f bf = *(const v2f*)&sWeT[c * LSTRIDE + k0];
        acc = wmma_f32_4(a, bf, acc);
      }
      // per-lane: 8 (j,c) elements of edge_x; relu + masked aggregation
      float partial = 0.f;
#pragma unroll
      for (int v = 0; v < 8; ++v) {
        int jl = mt * 16 + v + 8 * half;
        float ex = acc[v];
        rstash[t][v] = fmaxf(ex, 0.f);
        partial += sMask[jl] * ex * sN[jl * LSTRIDE + c];
      }
      atomicAdd(&sAgg[c], partial);           // ds_add_f32
    }
    __syncthreads();                          // all LDS reads of sE done

    // Write relu(edge_x) back into sE as [j][c], then coalesced B128 store
#pragma unroll
    for (int t = 0; t < 2; ++t) {
      int tile = wave * 2 + t;
      int mt = tile >> 2, nt = tile & 3;
      int c = nt * 16 + lidx;
#pragma unroll
      for (int v = 0; v < 8; ++v) {
        int jl = mt * 16 + v + 8 * half;
        sE[jl * LSTRIDE + c] = rstash[t][v];
      }
    }
    __syncthreads();
#pragma unroll
    for (int e = 0; e < 4; ++e) {
      int idx = (tid + e * 256) * 4;
      int j = idx >> 6, k = idx & 63;
      *(float4*)(edge_out + rowbase + (size_t)jbase * DD + idx) =
          *(float4*)&sE[j * LSTRIDE + k];
    }
    __syncthreads();
  }

  if (tid < 64) {
    node_out[(size_t)bi * DD + tid] =
        fmaxf(sAgg[tid], 0.f) + relu_sx[(size_t)bi * DD + tid];
  }
}

// ---------------------------------------------------------------------------
extern "C" void kernel_launch(void* const* d_in, const int* in_sizes, int n_in,
                              void* d_out, int out_size, void* d_ws, size_t ws_size,
                              hipStream_t stream) {
  const float* Amat     = (const float*)d_in[0];
  const float* emb_node = (const float*)d_in[1];
  const float* emb_edge = (const float*)d_in[2];
  const float* Wn       = (const float*)d_in[3];
  const float* bn       = (const float*)d_in[4];
  const float* Wsn      = (const float*)d_in[5];
  const float* bsn      = (const float*)d_in[6];
  const float* We       = (const float*)d_in[7];
  const float* be       = (const float*)d_in[8];

  float* node_out = (float*)d_out;                        // [8,256,64]
  float* edge_out = node_out + (size_t)BB * NN * DD;      // [8,256,256,64]

  float* node_x  = (float*)d_ws;                          // [2048,64] f32
  float* relu_sx = node_x + (size_t)BB * NN * DD;         // [2048,64] f32

  node_gemm_kernel<<<(BB * NN) / 64, 256, 0, stream>>>(
      emb_node, Wn, bn, Wsn, bsn, node_x, relu_sx);
  edge_kernel<<<BB * NN, 256, 0, stream>>>(
      Amat, emb_edge, We, be, node_x, relu_sx, node_out, edge_out);
}